// GCNStacker_24215025615566
// MI455X (gfx1250) — compile-verified
//
#include <hip/hip_runtime.h>
#include <hip/hip_bf16.h>
#include <math.h>

// ---------------- problem constants ----------------
#define NT_TERMS 26000
#define NE_EDGES 120000
#define NBATCH   64
#define NFEAT    14
#define GOEMB    8
#define HID      64
#define C0PAD    24     // 22 real input channels padded to mult-of-4
#define C0REAL   22
#define BC       8      // batch chunk size
#define NCHUNK   (NBATCH / BC)

typedef float v2f __attribute__((ext_vector_type(2)));
typedef float v8f __attribute__((ext_vector_type(8)));

// ---------------- utility kernels ----------------
__global__ void zero_f32(float* __restrict__ p, long n) {
    long i = (long)blockIdx.x * blockDim.x + threadIdx.x;
    long stride = (long)gridDim.x * blockDim.x;
    for (; i < n; i += stride) p[i] = 0.0f;
}

__global__ void deg_count(const int* __restrict__ rowIdx, float* __restrict__ deg, int nE) {
    int i = blockIdx.x * blockDim.x + threadIdx.x;
    if (i < nE) atomicAdd(&deg[rowIdx[i]], 1.0f);
}

__global__ void deg_invert(float* __restrict__ deg, int n) {
    int i = blockIdx.x * blockDim.x + threadIdx.x;
    if (i < n) {
        float v = deg[i];
        deg[i] = 1.0f / (v < 1.0f ? 1.0f : v);
    }
}

// Build h0[n, b, c] = concat(x[bBase+b, n, 0:14], emb[term_idx[n], 0:8], 0, 0)
__global__ void build_h0(const float* __restrict__ x,
                         const int* __restrict__ term_idx,
                         const float* __restrict__ emb,
                         float* __restrict__ h0,
                         int bBase) {
    long i = (long)blockIdx.x * blockDim.x + threadIdx.x;
    const long total = (long)NT_TERMS * BC * C0PAD;
    if (i >= total) return;
    int c  = (int)(i % C0PAD);
    long nb = i / C0PAD;
    int b  = (int)(nb % BC);
    int n  = (int)(nb / BC);
    float v;
    if (c < NFEAT) {
        v = x[((long)(bBase + b) * NT_TERMS + n) * NFEAT + c];
    } else if (c < NFEAT + GOEMB) {
        v = emb[(long)term_idx[n] * GOEMB + (c - NFEAT)];
    } else {
        v = 0.0f;   // zero pad channels 22,23
    }
    h0[i] = v;
}

// One wave per edge: agg[row, :, :] += hin[col, :, :]  (span = BC*CIN contiguous floats)
__global__ void scatter_edges(const float* __restrict__ hin,
                              float* __restrict__ agg,
                              const int* __restrict__ rowIdx,
                              const int* __restrict__ colIdx,
                              int nE, int span) {
    int wave = (blockIdx.x * blockDim.x + threadIdx.x) >> 5;
    int lane = threadIdx.x & 31;
    if (wave >= nE) return;
    int r = rowIdx[wave];
    int c = colIdx[wave];
    const float* src = hin + (long)c * span;
    float*       dst = agg + (long)r * span;
    for (int j = lane; j < span; j += 32)
        atomicAdd(dst + j, src[j]);
}

// ---------------- fused GCN layer GEMM (f32 WMMA, W staged in LDS) ----------------
// out[m, o] = relu( bias[o] + sum_k (hin[m,k] + deginv[m/BC]*agg[m,k]) * W[o,k] )
// One wave handles one 16-row M tile and all COUT columns (NTILE 16-col tiles).
// A frag (16x4 f32): lane<16 -> M=lane, K={k,k+1}; lane>=16 -> M=lane-16, K={k+2,k+3}
// B frag (4x16 f32): lane<16 -> N=lane, K={k,k+1}; lane>=16 -> N=lane-16, K={k+2,k+3}
// D (16x16 f32, 8 VGPRs): vgpr r -> M = r + 8*(lane/16), N = lane%16
template <int CIN, int CREAL, int COUT>
__global__ __launch_bounds__(256) void gemm_gcn(const float* __restrict__ hin,
                                                const float* __restrict__ agg,
                                                const float* __restrict__ deginv,
                                                const float* __restrict__ W,     // [COUT, CREAL]
                                                const float* __restrict__ bias,  // [COUT]
                                                float* __restrict__ out,         // [M, COUT]
                                                int M) {
    constexpr int NTILE = COUT / 16;
    __shared__ float Wlds[COUT * CIN];          // zero-padded [COUT, CIN], K-major rows

    // cooperative fill: whole block stages W once (zero-pads CREAL..CIN-1)
    for (int idx = threadIdx.x; idx < COUT * CIN; idx += 256) {
        int o = idx / CIN;
        int k = idx % CIN;
        Wlds[idx] = (k < CREAL) ? W[(long)o * CREAL + k] : 0.0f;
    }
    __syncthreads();

    int wave = (blockIdx.x * blockDim.x + threadIdx.x) >> 5;
    int lane = threadIdx.x & 31;
    int m0 = wave * 16;
    if (m0 >= M) return;
    int half = lane >> 4;     // 0 or 1
    int l15  = lane & 15;

    int rowA = m0 + l15;                       // row this lane loads for A
    float dinv = deginv[rowA >> 3];            // BC == 8
    const float* arow = hin + (long)rowA * CIN;
    const float* grow = agg + (long)rowA * CIN;

    v8f zero = {};
    v8f acc[NTILE];
#pragma unroll
    for (int t = 0; t < NTILE; ++t) acc[t] = zero;

#pragma unroll
    for (int k = 0; k < CIN; k += 4) {
        int ka = k + 2 * half;
        v2f a;
        a.x = arow[ka]     + dinv * grow[ka];
        a.y = arow[ka + 1] + dinv * grow[ka + 1];
#pragma unroll
        for (int t = 0; t < NTILE; ++t) {
            int ncol = t * 16 + l15;
            const v2f* wrow = (const v2f*)&Wlds[ncol * CIN + ka];  // ka even -> 8B aligned
            v2f bf = *wrow;                                        // ds_load_b64
            acc[t] = __builtin_amdgcn_wmma_f32_16x16x4_f32(
                false, a, false, bf, (short)0, acc[t], false, false);
        }
    }

    // epilogue: bias + relu, store D
#pragma unroll
    for (int t = 0; t < NTILE; ++t) {
        int ncol = t * 16 + l15;
        float bv = bias[ncol];
#pragma unroll
        for (int r = 0; r < 8; ++r) {
            int m = m0 + r + 8 * half;
            float v = acc[t][r] + bv;
            out[(long)m * COUT + ncol] = v > 0.0f ? v : 0.0f;
        }
    }
}

// Final head: out[bBase+b, n] = sigmoid(dot(h3[n,b,:], Wout) + bout)
__global__ void head_out(const float* __restrict__ h3,   // [N, BC, 32]
                         const float* __restrict__ Wout, // [32]
                         const float* __restrict__ bout, // [1]
                         float* __restrict__ out,        // [B, N]
                         int bBase) {
    int i = blockIdx.x * blockDim.x + threadIdx.x;    // over N*BC
    if (i >= NT_TERMS * BC) return;
    int b = i % BC;
    int n = i / BC;
    const float* h = h3 + (long)i * (HID / 2);
    float s = bout[0];
#pragma unroll
    for (int c = 0; c < HID / 2; ++c) s += h[c] * Wout[c];
    out[(long)(bBase + b) * NT_TERMS + n] = 1.0f / (1.0f + __expf(-s));
}

// ---------------- host launcher ----------------
extern "C" void kernel_launch(void* const* d_in, const int* in_sizes, int n_in,
                              void* d_out, int out_size, void* d_ws, size_t ws_size,
                              hipStream_t stream) {
    const float* x        = (const float*)d_in[0];   // [B, N, 14]
    const int*   eidx     = (const int*)  d_in[1];   // [2, E]
    const int*   term_idx = (const int*)  d_in[2];   // [N]
    const float* emb      = (const float*)d_in[3];   // [N, 8]
    const float* W1 = (const float*)d_in[4];  const float* b1 = (const float*)d_in[5];
    const float* W2 = (const float*)d_in[6];  const float* b2 = (const float*)d_in[7];
    const float* W3 = (const float*)d_in[8];  const float* b3 = (const float*)d_in[9];
    const float* Wo = (const float*)d_in[10]; const float* bo = (const float*)d_in[11];
    float* out = (float*)d_out;

    const int* rowIdx = eidx;             // edge_index[0]
    const int* colIdx = eidx + NE_EDGES;  // edge_index[1]

    // workspace carve-up (floats)
    float* ws    = (float*)d_ws;
    const long NB   = (long)NT_TERMS * BC;
    float* deginv = ws;                     // N
    float* h0  = deginv + NT_TERMS;         // N*BC*24
    float* hA  = h0  + NB * C0PAD;          // N*BC*64
    float* hB  = hA  + NB * HID;            // N*BC*64
    float* agg = hB  + NB * HID;            // N*BC*64
    float* h3  = agg + NB * HID;            // N*BC*32

    const int M = (int)NB;                  // 208000 rows per chunk (mult of 16)
    const int gemmBlocks = (M / 16 + 7) / 8;        // 8 waves (M-tiles) per 256-thread block
    const int scatBlocks = (NE_EDGES + 7) / 8;      // 8 edges per 256-thread block

    // degree -> inverse degree (once)
    zero_f32<<<256, 256, 0, stream>>>(deginv, NT_TERMS);
    deg_count<<<(NE_EDGES + 255) / 256, 256, 0, stream>>>(rowIdx, deginv, NE_EDGES);
    deg_invert<<<(NT_TERMS + 255) / 256, 256, 0, stream>>>(deginv, NT_TERMS);

    for (int chunk = 0; chunk < NCHUNK; ++chunk) {
        int bBase = chunk * BC;

        // layer 0 input: concat + pad
        long h0elems = NB * C0PAD;
        build_h0<<<(int)((h0elems + 255) / 256), 256, 0, stream>>>(x, term_idx, emb, h0, bBase);

        // layer 1: 24(->22 real) -> 64
        zero_f32<<<2048, 256, 0, stream>>>(agg, NB * C0PAD);
        scatter_edges<<<scatBlocks, 256, 0, stream>>>(h0, agg, rowIdx, colIdx, NE_EDGES, BC * C0PAD);
        gemm_gcn<C0PAD, C0REAL, HID><<<gemmBlocks, 256, 0, stream>>>(h0, agg, deginv, W1, b1, hA, M);

        // layer 2: 64 -> 64
        zero_f32<<<4096, 256, 0, stream>>>(agg, NB * HID);
        scatter_edges<<<scatBlocks, 256, 0, stream>>>(hA, agg, rowIdx, colIdx, NE_EDGES, BC * HID);
        gemm_gcn<HID, HID, HID><<<gemmBlocks, 256, 0, stream>>>(hA, agg, deginv, W2, b2, hB, M);

        // layer 3: 64 -> 32
        zero_f32<<<4096, 256, 0, stream>>>(agg, NB * HID);
        scatter_edges<<<scatBlocks, 256, 0, stream>>>(hB, agg, rowIdx, colIdx, NE_EDGES, BC * HID);
        gemm_gcn<HID, HID, HID / 2><<<gemmBlocks, 256, 0, stream>>>(hB, agg, deginv, W3, b3, h3, M);

        // sigmoid head
        head_out<<<(NT_TERMS * BC + 255) / 256, 256, 0, stream>>>(h3, Wo, bo, out, bBase);
    }
}